// Pooling_27745488732840
// MI455X (gfx1250) — compile-verified
//
#include <hip/hip_runtime.h>

#define N_AGENTS 1024
#define HIDDEN   128
#define FGRID    32
#define POOLED   2048   // HIDDEN * 4 * 4

typedef __attribute__((ext_vector_type(2))) float v2f;
typedef __attribute__((ext_vector_type(4))) float v4f;
typedef __attribute__((ext_vector_type(8))) float v8f;

// ---------------------------------------------------------------------------
// Kernel A: per-agent social pooling (scatter + view-bug reindex + 8x8 pool)
// One block per agent. owner[] built with LDS atomicMax (last-writer-wins ==
// max j, matching the fancy-assignment scatter order). Each owned cell's 128
// hidden values collapse into 16 pooled slots (8 contiguous d per slot):
//   flat = cell*128 + d ; c = cell>>3 ; y = (cell&7)*4 + d/32 ; x = d&31
//   slot = c*16 + (y>>3)*4 + (x>>3)
// ---------------------------------------------------------------------------
__global__ __launch_bounds__(256) void social_pool_kernel(
    const float* __restrict__ hidden,   // (1024,128)
    const float* __restrict__ obs2,     // (1024,2)
    float* __restrict__ pooled)         // (1024,2048) workspace
{
  __shared__ int   owner[FGRID * FGRID];   // 4 KB
  __shared__ float pgrid[POOLED];          // 8 KB
  const int i   = blockIdx.x;
  const int tid = threadIdx.x;

  for (int t = tid; t < FGRID * FGRID; t += 256) owner[t] = -1;
  for (int t = tid; t < POOLED; t += 256) pgrid[t] = 0.0f;

  const float xi = obs2[2 * i + 0];
  const float yi = obs2[2 * i + 1];
  __syncthreads();

  // Pairwise scatter: rel = (obs2[j]-obs2[i])*POOL/CELL_SIDE + GRID/2.
  // NaN in any coordinate fails the range compares (matches the NaN masks).
  for (int j = tid; j < N_AGENTS; j += 256) {
    float xj = obs2[2 * j + 0];
    float yj = obs2[2 * j + 1];
    float rx = (xj - xi) * 8.0f + 16.0f;
    float ry = (yj - yi) * 8.0f + 16.0f;
    bool valid = (j != i) && (rx >= 0.0f) && (rx < 32.0f)
                          && (ry >= 0.0f) && (ry < 32.0f);
    if (valid) {
      int cell = (int)rx * FGRID + (int)ry;   // trunc toward zero == .long()
      atomicMax(&owner[cell], j);             // ds_max_rtn on LDS
    }
  }
  __syncthreads();

  // Accumulate owned cells into the pooled grid (ds_add_f32 on LDS).
  for (int cell = tid; cell < FGRID * FGRID; cell += 256) {
    int j = owner[cell];
    if (j < 0) continue;
    const float* h = hidden + (size_t)j * HIDDEN;
    const int c     = cell >> 3;
    const int ybase = (cell & 7) * 4;
#pragma unroll
    for (int g = 0; g < 16; ++g) {            // 8 consecutive d per slot
      const int d0 = g * 8;
      v4f h0 = *(const v4f*)(h + d0);
      v4f h1 = *(const v4f*)(h + d0 + 4);
      float s = (h0.x + h0.y + h0.z + h0.w) + (h1.x + h1.y + h1.z + h1.w);
      const int y = ybase + (d0 >> 5);
      atomicAdd(&pgrid[c * 16 + (y >> 3) * 4 + (g & 3)], s);
    }
  }
  __syncthreads();

  // Flush pooled feature to workspace (vectorized b128 stores).
  v4f* dst = (v4f*)(pooled + (size_t)i * POOLED);
  for (int t = tid; t < POOLED / 4; t += 256) dst[t] = ((const v4f*)pgrid)[t];
}

// ---------------------------------------------------------------------------
// Kernel B: out(1024x128) = pooled(1024x2048) @ W^T(2048x128) + b
// fp32 WMMA path (V_WMMA_F32_16X16X4_F32): keeps reference fp32 precision;
// problem is bandwidth-bound so the fp32 matrix pipe is the right choice.
// Each wave computes one 16x16 C tile; 8 waves/block cover all 8 n-tiles of
// one 16-row strip; 64 blocks cover M=1024.
//
// A 16x4 f32 fragment: lanes 0-15 -> m=lane, k in {k0,k0+1};
//                      lanes 16-31 -> m=lane-16, k in {k0+2,k0+3}. (ISA 7.12.2)
// B 4x16 f32 fragment mirrors with n in place of m. Both are contiguous
// float2 loads per lane.
// ---------------------------------------------------------------------------
__global__ __launch_bounds__(256) void gemm_wmma_kernel(
    const float* __restrict__ pooled,   // (1024,2048)
    const float* __restrict__ W,        // (128,2048)
    const float* __restrict__ bias,     // (128)
    float* __restrict__ out)            // (1024,128)
{
  const int lane  = threadIdx.x & 31;
  const int wave  = threadIdx.x >> 5;
  const int l16   = lane & 15;
  const int khalf = lane >> 4;          // selects k-pair {0,1} vs {2,3}
  const int m0 = blockIdx.x * 16;
  const int n0 = wave * 16;

  const float* Ap = pooled + (size_t)(m0 + l16) * POOLED + khalf * 2;
  const float* Bp = W      + (size_t)(n0 + l16) * POOLED + khalf * 2;

  v8f acc = {};
#pragma unroll 8
  for (int k0 = 0; k0 < POOLED; k0 += 4) {
    v2f a = *(const v2f*)(Ap + k0);
    v2f b = *(const v2f*)(Bp + k0);
    // (neg_a, A, neg_b, B, c_mod, C, reuse_a, reuse_b)
    acc = __builtin_amdgcn_wmma_f32_16x16x4_f32(
        false, a, false, b, (short)0, acc, false, false);
  }

  // C/D layout: lane-half selects M offset 8; VGPR r -> row m0 + half*8 + r;
  // column n = n0 + l16.
  const float bv   = bias[n0 + l16];
  const int   mrow = m0 + khalf * 8;
#pragma unroll
  for (int r = 0; r < 8; ++r) {
    out[(size_t)(mrow + r) * HIDDEN + n0 + l16] = acc[r] + bv;
  }
}

extern "C" void kernel_launch(void* const* d_in, const int* in_sizes, int n_in,
                              void* d_out, int out_size, void* d_ws, size_t ws_size,
                              hipStream_t stream) {
  const float* hidden = (const float*)d_in[0];
  // d_in[1] = obs1 (unused by the reference computation)
  const float* obs2 = (const float*)d_in[2];
  const float* W    = (const float*)d_in[3];
  const float* bias = (const float*)d_in[4];
  float* out    = (float*)d_out;
  float* pooled = (float*)d_ws;   // needs 1024*2048*4 = 8 MB scratch

  social_pool_kernel<<<N_AGENTS, 256, 0, stream>>>(hidden, obs2, pooled);
  gemm_wmma_kernel<<<N_AGENTS / 16, 256, 0, stream>>>(pooled, W, bias, out);
}